// MambaLayer_16947940950326
// MI455X (gfx1250) — compile-verified
//
#include <hip/hip_runtime.h>
#include <cstdint>
#include <cstddef>

// ---------------------------------------------------------------------------
// Mamba layer forward for MI455X (gfx1250, wave32).
// All GEMMs run on V_WMMA_F32_16X16X4_F32 (fp32 in / fp32 acc -> exact wrt ref),
// register-blocked MTxNT tiles per wave to raise FLOP/byte out of L2.
// ---------------------------------------------------------------------------

typedef __attribute__((ext_vector_type(2))) float v2f;
typedef __attribute__((ext_vector_type(8))) float v8f;

#define DEV __device__ __forceinline__

constexpr int BATCH   = 2;
constexpr int SEQ     = 2048;
constexpr int DMODEL  = 1024;
constexpr int DIN     = 2048;                 // d_inner
constexpr int DSTATE  = 16;
constexpr int DTRANK  = 64;
constexpr int MTOT    = BATCH * SEQ;          // 4096 GEMM rows
constexpr int XPROJ_N = DTRANK + 2 * DSTATE;  // 96

// -------------------------- WMMA f32 GEMM core -----------------------------
// acc[MT][NT] (16x16 tiles) += A[tile_m.., :K] * W[tile_n.., :K]^T (row-major).
//
// f32 A 16x4 fragment layout (ISA 7.12.2): lanes 0-15 -> M rows; the VGPR pair
// holds K={0,1} for lanes 0-15 and K={2,3} for lanes 16-31 => per lane the two
// values are CONTIGUOUS in a row-major row -> single float2 (b64) load.
// B (KxN = W^T tile) mirrors this from the row-major W[N,K].
template <int MT, int NT>
DEV void wmma_gemm_block(const float* __restrict__ A, const float* __restrict__ W,
                         int lda, int ldw, int tile_m, int tile_n, int K,
                         v8f (&acc)[MT][NT]) {
  const int lane = threadIdx.x & 31;
  const int half = lane >> 4;     // K sub-pair selector
  const int lr   = lane & 15;     // M (for A) / N (for B) within tile

  const float* ap[MT];
  const float* bp[NT];
#pragma unroll
  for (int i = 0; i < MT; ++i)
    ap[i] = A + (size_t)(tile_m + i * 16 + lr) * lda + 2 * half;
#pragma unroll
  for (int j = 0; j < NT; ++j)
    bp[j] = W + (size_t)(tile_n + j * 16 + lr) * ldw + 2 * half;

  const v8f zero = {};
#pragma unroll
  for (int i = 0; i < MT; ++i)
#pragma unroll
    for (int j = 0; j < NT; ++j) acc[i][j] = zero;

  // MT+NT loads feed MT*NT WMMAs per K-step; MT*NT independent acc chains.
  for (int k = 0; k < K; k += 4) {
    v2f a[MT], b[NT];
#pragma unroll
    for (int i = 0; i < MT; ++i) a[i] = *(const v2f*)(ap[i] + k);
#pragma unroll
    for (int j = 0; j < NT; ++j) b[j] = *(const v2f*)(bp[j] + k);
#pragma unroll
    for (int i = 0; i < MT; ++i)
#pragma unroll
      for (int j = 0; j < NT; ++j)
        acc[i][j] = __builtin_amdgcn_wmma_f32_16x16x4_f32(
            false, a[i], false, b[j], (short)0, acc[i][j], false, false);
  }
}

// ------------------------------ kernels ------------------------------------
// Big-GEMM launch geometry: 256 threads = 8 waves as 4(M) x 2(N);
// wave tile 64x32 -> block tile 256(M) x 64(N).

// xz = x @ W_in^T ; split columns into xs (first DIN) and z (last DIN).
__global__ void k_in_proj(const float* __restrict__ x, const float* __restrict__ Win,
                          float* __restrict__ xs, float* __restrict__ z) {
  const int wave   = threadIdx.x >> 5;
  const int tile_m = blockIdx.y * 256 + (wave >> 1) * 64;
  const int tile_n = blockIdx.x * 64 + (wave & 1) * 32;
  v8f acc[4][2];
  wmma_gemm_block<4, 2>(x, Win, DMODEL, DMODEL, tile_m, tile_n, DMODEL, acc);
  const int lane = threadIdx.x & 31;
#pragma unroll
  for (int i = 0; i < 4; ++i)
#pragma unroll
    for (int j = 0; j < 2; ++j) {
      const int n     = tile_n + j * 16 + (lane & 15);
      const int mbase = tile_m + i * 16 + (lane >> 4) * 8;
      float* dst = (n < DIN) ? xs : z;
      const int nn = (n < DIN) ? n : n - DIN;
#pragma unroll
      for (int r = 0; r < 8; ++r)
        dst[(size_t)(mbase + r) * DIN + nn] = acc[i][j][r];
    }
}

// Causal depthwise conv1d (k=4) + bias + SiLU.
__global__ void k_conv_silu(const float* __restrict__ xs, const float* __restrict__ cw,
                            const float* __restrict__ cb, float* __restrict__ xc) {
  const size_t idx = (size_t)blockIdx.x * blockDim.x + threadIdx.x;
  const int d = (int)(idx & (DIN - 1));
  const int l = (int)((idx >> 11) & (SEQ - 1));
  const float* w = cw + d * 4;                      // conv_w[d][0][0..3]
  const float* col = xs + (idx - (size_t)l * DIN);  // (b, l=0, d)
  float s = cb[d];
#pragma unroll
  for (int j = 0; j < 4; ++j) {
    const int ll = l - 3 + j;
    if (ll >= 0) s += w[j] * col[(size_t)ll * DIN];
  }
  xc[idx] = s / (1.0f + __expf(-s));                // SiLU
}

// x_dbl = xc @ W_xproj^T   (N = 96 -> NT=1, waves stacked in M: 512x16 block)
__global__ void k_xproj(const float* __restrict__ xc, const float* __restrict__ Wxp,
                        float* __restrict__ xdbl) {
  const int wave   = threadIdx.x >> 5;
  const int tile_m = blockIdx.y * 512 + wave * 64;
  const int tile_n = blockIdx.x * 16;
  v8f acc[4][1];
  wmma_gemm_block<4, 1>(xc, Wxp, DIN, DIN, tile_m, tile_n, DIN, acc);
  const int lane = threadIdx.x & 31;
  const int n = tile_n + (lane & 15);
#pragma unroll
  for (int i = 0; i < 4; ++i) {
    const int mbase = tile_m + i * 16 + (lane >> 4) * 8;
#pragma unroll
    for (int r = 0; r < 8; ++r)
      xdbl[(size_t)(mbase + r) * XPROJ_N + n] = acc[i][0][r];
  }
}

// dt = softplus(x_dbl[:, :64] @ W_dt^T + b_dt)   (K = 64, lda = 96)
__global__ void k_dt(const float* __restrict__ xdbl, const float* __restrict__ Wdt,
                     const float* __restrict__ bdt, float* __restrict__ dt) {
  const int wave   = threadIdx.x >> 5;
  const int tile_m = blockIdx.y * 256 + (wave >> 1) * 64;
  const int tile_n = blockIdx.x * 64 + (wave & 1) * 32;
  v8f acc[4][2];
  wmma_gemm_block<4, 2>(xdbl, Wdt, XPROJ_N, DTRANK, tile_m, tile_n, DTRANK, acc);
  const int lane = threadIdx.x & 31;
#pragma unroll
  for (int i = 0; i < 4; ++i)
#pragma unroll
    for (int j = 0; j < 2; ++j) {
      const int n     = tile_n + j * 16 + (lane & 15);
      const int mbase = tile_m + i * 16 + (lane >> 4) * 8;
      const float bias = bdt[n];
#pragma unroll
      for (int r = 0; r < 8; ++r) {
        float v = acc[i][j][r] + bias;
        v = (v > 20.0f) ? v : __logf(1.0f + __expf(v));   // softplus
        dt[(size_t)(mbase + r) * DIN + n] = v;
      }
    }
}

// Selective scan: one thread per (b, d) channel, 16 states in registers.
// Fuses the D*x skip term and the SiLU(z) gate.
__global__ void k_scan(const float* __restrict__ xc, const float* __restrict__ z,
                       const float* __restrict__ dt, const float* __restrict__ xdbl,
                       const float* __restrict__ Alog, const float* __restrict__ Dp,
                       float* __restrict__ y) {
  const int tid = blockIdx.x * blockDim.x + threadIdx.x;  // 0..4095
  const int b = tid >> 11;
  const int d = tid & (DIN - 1);

  float A[DSTATE], h[DSTATE];
#pragma unroll
  for (int n = 0; n < DSTATE; ++n) {
    A[n] = -__expf(Alog[d * DSTATE + n]);
    h[n] = 0.0f;
  }
  const float Dd = Dp[d];
  const size_t chan = (size_t)b * SEQ * DIN + d;
  const float* dtp = dt + chan;
  const float* xp  = xc + chan;
  const float* zp  = z  + chan;
  const float* bcp = xdbl + (size_t)b * SEQ * XPROJ_N;
  float* yp = y + chan;

  for (int l = 0; l < SEQ; ++l) {
    const float dtv = dtp[(size_t)l * DIN];
    const float xv  = xp[(size_t)l * DIN];
    const float du  = dtv * xv;
    const float* Bv = bcp + (size_t)l * XPROJ_N + DTRANK;  // B row (16)
    const float* Cv = Bv + DSTATE;                          // C row (16)
    float acc = 0.0f;
#pragma unroll
    for (int n = 0; n < DSTATE; ++n) {
      h[n] = h[n] * __expf(dtv * A[n]) + du * Bv[n];
      acc += h[n] * Cv[n];
    }
    acc += xv * Dd;
    const float zv = zp[(size_t)l * DIN];
    acc *= zv / (1.0f + __expf(-zv));                       // * SiLU(z)
    yp[(size_t)l * DIN] = acc;
  }
}

// out = y @ W_out^T
__global__ void k_out_proj(const float* __restrict__ y, const float* __restrict__ Wout,
                           float* __restrict__ out) {
  const int wave   = threadIdx.x >> 5;
  const int tile_m = blockIdx.y * 256 + (wave >> 1) * 64;
  const int tile_n = blockIdx.x * 64 + (wave & 1) * 32;
  v8f acc[4][2];
  wmma_gemm_block<4, 2>(y, Wout, DIN, DIN, tile_m, tile_n, DIN, acc);
  const int lane = threadIdx.x & 31;
#pragma unroll
  for (int i = 0; i < 4; ++i)
#pragma unroll
    for (int j = 0; j < 2; ++j) {
      const int n     = tile_n + j * 16 + (lane & 15);
      const int mbase = tile_m + i * 16 + (lane >> 4) * 8;
#pragma unroll
      for (int r = 0; r < 8; ++r)
        out[(size_t)(mbase + r) * DMODEL + n] = acc[i][j][r];
    }
}

// ------------------------------ launcher -----------------------------------

extern "C" void kernel_launch(void* const* d_in, const int* in_sizes, int n_in,
                              void* d_out, int out_size, void* d_ws, size_t ws_size,
                              hipStream_t stream) {
  const float* x      = (const float*)d_in[0];
  const float* W_in   = (const float*)d_in[1];
  const float* conv_w = (const float*)d_in[2];
  const float* conv_b = (const float*)d_in[3];
  const float* W_xp   = (const float*)d_in[4];
  const float* W_dt   = (const float*)d_in[5];
  const float* b_dt   = (const float*)d_in[6];
  const float* A_log  = (const float*)d_in[7];
  const float* Dvec   = (const float*)d_in[8];
  const float* W_out  = (const float*)d_in[9];
  float* out = (float*)d_out;

  // Workspace layout (floats). Entire set (~130 MB) fits in the 192 MB L2.
  constexpr size_t NXS = (size_t)BATCH * SEQ * DIN;          // 8 Mi floats
  float* ws      = (float*)d_ws;
  float* xs_raw  = ws;                    // pre-conv xs      [B,L,DIN]
  float* z_buf   = ws + NXS;              // gate z           [B,L,DIN]
  float* xc_buf  = ws + 2 * NXS;          // conv+SiLU out    [B,L,DIN]
  float* xdbl    = ws + 3 * NXS;          // (dt_low|B|C)     [B,L,96]
  float* dt_buf  = ws + 3 * NXS + (size_t)MTOT * XPROJ_N;    // [B,L,DIN]
  float* y_buf   = xs_raw;                // raw xs dead after conv -> reuse

  const dim3 blk(256);  // 8 waves

  // 1) in_proj: [4096,1024] x [1024,4096]; block tile 256x64
  k_in_proj<<<dim3((2 * DIN) / 64, MTOT / 256), blk, 0, stream>>>(x, W_in, xs_raw, z_buf);

  // 2) causal depthwise conv + SiLU
  k_conv_silu<<<dim3((unsigned)(NXS / 256)), blk, 0, stream>>>(xs_raw, conv_w, conv_b, xc_buf);

  // 3) x_proj: [4096,2048] x [2048,96]; block tile 512x16
  k_xproj<<<dim3(XPROJ_N / 16, MTOT / 512), blk, 0, stream>>>(xc_buf, W_xp, xdbl);

  // 4) dt: [4096,64] x [64,2048] + softplus; block tile 256x64
  k_dt<<<dim3(DIN / 64, MTOT / 256), blk, 0, stream>>>(xdbl, W_dt, b_dt, dt_buf);

  // 5) selective scan (+ skip + gate): 4096 channels
  k_scan<<<dim3((BATCH * DIN) / 256), blk, 0, stream>>>(xc_buf, z_buf, dt_buf, xdbl,
                                                        A_log, Dvec, y_buf);

  // 6) out_proj: [4096,2048] x [2048,1024]; block tile 256x64
  k_out_proj<<<dim3(DMODEL / 64, MTOT / 256), blk, 0, stream>>>(y_buf, W_out, out);

  (void)in_sizes; (void)n_in; (void)out_size; (void)ws_size;
}